// CapsuleLayer_44066364457157
// MI455X (gfx1250) — compile-verified
//
#include <hip/hip_runtime.h>
#include <stdint.h>

// ---------------------------------------------------------------------------
// Capsule routing, fused for MI455X (gfx1250), full-fp32 path.
//   u_hat GEMM: per (n,j): [16d x 8p] @ [8p x 16b] = two chained
//   v_wmma_f32_16x16x4_f32 (all 32 lanes carry real data, no padding waste).
//   u_hat slab staged in 168KB LDS, bank-conflict-free [j][n][q][b] float4
//   layout; never materialized in HBM.
//   b_i = (sum_{k<i} out_k) . u_hat  => no logit storage, 3 sweeps total.
// ---------------------------------------------------------------------------

#define Bn 128
#define Jn 2048
#define Pn 8
#define Nn 32
#define Dn 16
#define JC 4                  // j's per chunk staged in LDS
#define JGROUPS 32            // blocks along J
#define CHUNKS_PER_BLOCK 16   // (Jn/JC)/JGROUPS
#define EPSF 1e-7f

typedef float v2f __attribute__((ext_vector_type(2)));
typedef float v8f __attribute__((ext_vector_type(8)));

// -------------------------- init kernel ------------------------------------
__global__ void k_zero_f32(float* __restrict__ p, int n) {
  int i = blockIdx.x * blockDim.x + threadIdx.x;
  if (i < n) p[i] = 0.f;
}

// ------------------------------- sweep kernel ------------------------------
// grid: (JGROUPS, Bn/16); block: 256 threads (8 wave32).
// W: f32 [N][J][D][P]; X: f32 [B][J][P]; Osum: f32 [B][N][D]
// Spart: f32 [JGROUPS][B][N][D]  (per-jgroup partial, exclusive ownership)
__global__ __launch_bounds__(256)
void k_sweep(const float* __restrict__ W,
             const float* __restrict__ X,
             const float* __restrict__ Osum,
             float* __restrict__ Spart) {
  // u_hat[j][n][d4-group q][b] : float4 holds d=4q..4q+3 for batch-col b.
  __shared__ __align__(16) float4 uhat4[JC][Nn][4][16];   // 128 KB
  __shared__ float ccoef[JC][Nn][16];                     //   8 KB
  __shared__ __align__(16) float4 ol4[Nn][4][16];         //  32 KB  O_sum slice

  const int t    = threadIdx.x;
  const int wave = t >> 5;
  const int lane = t & 31;
  const int lo16 = lane & 15;      // d-row (A) / b-col (B,D)
  const int hi   = lane >> 4;      // K-half selector for 16x16x4 layout
  const int jgrp = blockIdx.x;     // 0..31
  const int b0   = blockIdx.y * 16;

  // stage O_sum slice, transposed to [n][q][b] for conflict-free reads
  for (int idx = t; idx < 16 * Nn * Dn; idx += 256) {
    const int b = idx >> 9, n = (idx >> 4) & 31, d = idx & 15;
    ((float*)ol4)[((n * 4 + (d >> 2)) * 16 + b) * 4 + (d & 3)] =
        Osum[((size_t)(b0 + b) * Nn + n) * Dn + d];
  }

  float sacc[2][Dn];
#pragma unroll
  for (int it = 0; it < 2; ++it)
#pragma unroll
    for (int d = 0; d < Dn; ++d) sacc[it][d] = 0.f;

  const int jl_w  = wave & 3;          // wave's j within chunk
  const int nbase = (wave >> 2) * 16;  // wave's n half

  for (int c = 0; c < CHUNKS_PER_BLOCK; ++c) {
    const int j0 = (jgrp * CHUNKS_PER_BLOCK + c) * JC;
    __syncthreads();  // previous chunk's readers done before overwriting uhat

    // prefetch next chunk's W (128 (n,j) units x 512B)
    if (c + 1 < CHUNKS_PER_BLOCK && t < 128) {
      const int pn = t & 31, pj = (t >> 5) & 3;
      const float* pf = W + ((size_t)pn * Jn + (j0 + JC + pj)) * (Dn * Pn);
      __builtin_prefetch(pf, 0, 1);
      __builtin_prefetch(pf + 64, 0, 1);
    }

    // ---- stage 1: u_hat tiles via two chained f32 WMMAs -------------------
    {
      const int j = j0 + jl_w;
      // B operand (4x16): lane -> x[b0+lo16, j, k0 + 2*hi + {0,1}]
      const float* xp = X + ((size_t)(b0 + lo16) * Jn + j) * Pn + 2 * hi;
      const v2f xb0 = *(const v2f*)(xp);       // K 0..3
      const v2f xb1 = *(const v2f*)(xp + 4);   // K 4..7

      for (int nn = 0; nn < 16; ++nn) {
        const int n = nbase + nn;
        // A operand (16x4): lane -> W[n, j, d=lo16, k0 + 2*hi + {0,1}]
        const float* wp = W + (((size_t)n * Jn + j) * Dn + lo16) * Pn + 2 * hi;
        const v2f wa0 = *(const v2f*)(wp);
        const v2f wa1 = *(const v2f*)(wp + 4);

        v8f acc = {0.f, 0.f, 0.f, 0.f, 0.f, 0.f, 0.f, 0.f};
        acc = __builtin_amdgcn_wmma_f32_16x16x4_f32(
            false, wa0, false, xb0, (short)0, acc, false, false);
        acc = __builtin_amdgcn_wmma_f32_16x16x4_f32(
            false, wa1, false, xb1, (short)0, acc, false, false);

        // D layout: vgpr r -> d = 8*hi + r, b = lo16
        const int q0 = hi * 2;
        uhat4[jl_w][n][q0][lo16]     = make_float4(acc[0], acc[1], acc[2], acc[3]);
        uhat4[jl_w][n][q0 + 1][lo16] = make_float4(acc[4], acc[5], acc[6], acc[7]);
      }
    }
    __syncthreads();

    // ---- phase A: logits (O_sum . u_hat), softmax over n ------------------
    if (t < JC * 16) {
      const int jl = t >> 4, bb = t & 15;
      float lg[Nn];
      float mx = -3.0e38f;
      for (int n = 0; n < Nn; ++n) {
        float l = 0.f;
#pragma unroll
        for (int q = 0; q < 4; ++q) {
          const float4 u = uhat4[jl][n][q][bb];
          const float4 o = ol4[n][q][bb];
          l += o.x * u.x + o.y * u.y + o.z * u.z + o.w * u.w;
        }
        lg[n] = l;
        mx = fmaxf(mx, l);
      }
      float ssum = 0.f;
#pragma unroll
      for (int n = 0; n < Nn; ++n) { lg[n] = __expf(lg[n] - mx); ssum += lg[n]; }
      const float inv = 1.f / ssum;
#pragma unroll
      for (int n = 0; n < Nn; ++n) ccoef[jl][n][bb] = lg[n] * inv;
    }
    __syncthreads();

    // ---- phase B: s += c * u_hat, race-free (thread owns (n,b)) -----------
#pragma unroll
    for (int it = 0; it < 2; ++it) {
      const int idx = it * 256 + t;
      const int n = idx >> 4, bb = idx & 15;
#pragma unroll
      for (int jl = 0; jl < JC; ++jl) {
        const float cw = ccoef[jl][n][bb];
#pragma unroll
        for (int q = 0; q < 4; ++q) {
          const float4 u = uhat4[jl][n][q][bb];
          sacc[it][4 * q + 0] += cw * u.x;
          sacc[it][4 * q + 1] += cw * u.y;
          sacc[it][4 * q + 2] += cw * u.z;
          sacc[it][4 * q + 3] += cw * u.w;
        }
      }
    }
  }

  // write this jgroup's deterministic partial: Spart[jgrp][b][n][d]
#pragma unroll
  for (int it = 0; it < 2; ++it) {
    const int idx = it * 256 + t;
    const int n = idx >> 4, bb = idx & 15;
    float4* dst = (float4*)(Spart + (((size_t)jgrp * Bn + (b0 + bb)) * Nn + n) * Dn);
#pragma unroll
    for (int q = 0; q < 4; ++q)
      dst[q] = make_float4(sacc[it][4 * q], sacc[it][4 * q + 1],
                           sacc[it][4 * q + 2], sacc[it][4 * q + 3]);
  }
}

// ------------------------------ squash kernel ------------------------------
// 4096 threads: t -> (b, n). Reduce 32 partials, add bias, squash, update
// O_sum; write d_out on final iteration.
__global__ void k_squash(const float* __restrict__ Spart,
                         float* __restrict__ Osum,
                         const float* __restrict__ bias,
                         float* __restrict__ out, int is_last) {
  const int t = blockIdx.x * blockDim.x + threadIdx.x;
  if (t >= Bn * Nn) return;
  const int n = t & (Nn - 1);

  float s[Dn];
  {
    const float4* bp = (const float4*)(bias + (size_t)n * Dn);
#pragma unroll
    for (int q = 0; q < 4; ++q) {
      float4 bv = bp[q];
      s[4 * q + 0] = bv.x; s[4 * q + 1] = bv.y;
      s[4 * q + 2] = bv.z; s[4 * q + 3] = bv.w;
    }
  }
  for (int g = 0; g < JGROUPS; ++g) {
    const float4* sp = (const float4*)(Spart + ((size_t)g * Bn * Nn + t) * Dn);
#pragma unroll
    for (int q = 0; q < 4; ++q) {
      float4 sv = sp[q];
      s[4 * q + 0] += sv.x; s[4 * q + 1] += sv.y;
      s[4 * q + 2] += sv.z; s[4 * q + 3] += sv.w;
    }
  }
  float sq = 0.f;
#pragma unroll
  for (int d = 0; d < Dn; ++d) sq += s[d] * s[d];
  const float scale = sq / ((1.f + sq) * sqrtf(sq + EPSF));

  float* op = Osum + (size_t)t * Dn;
#pragma unroll
  for (int d = 0; d < Dn; ++d) {
    const float v = scale * s[d];
    op[d] += v;                       // running sum of outputs -> next logits
    if (is_last) out[(size_t)t * Dn + d] = v;
  }
}

// -------------------------------- launcher ---------------------------------
extern "C" void kernel_launch(void* const* d_in, const int* in_sizes, int n_in,
                              void* d_out, int out_size, void* d_ws, size_t ws_size,
                              hipStream_t stream) {
  const float* x    = (const float*)d_in[0];  // [B,J,P]
  const float* W    = (const float*)d_in[1];  // [N,J,D,P]
  const float* bias = (const float*)d_in[2];  // [N,D]
  float* out = (float*)d_out;                 // [B,N,D]

  char* ws = (char*)d_ws;
  float* Spart = (float*)(ws);                 // 8,388,608 B
  float* Osum  = (float*)(ws + 8388608);       //   262,144 B

  k_zero_f32<<<(Bn * Nn * Dn + 255) / 256, 256, 0, stream>>>(Osum, Bn * Nn * Dn);

  dim3 sg(JGROUPS, Bn / 16);
  for (int it = 0; it < 3; ++it) {
    k_sweep<<<sg, 256, 0, stream>>>(W, x, Osum, Spart);
    k_squash<<<(Bn * Nn + 255) / 256, 256, 0, stream>>>(Spart, Osum, bias, out,
                                                        it == 2 ? 1 : 0);
  }
}